// mHCGenerator_nStream_80951543595220
// MI455X (gfx1250) — compile-verified
//
#include <hip/hip_runtime.h>
#include <hip/hip_bf16.h>
#include <math.h>

// ---------------------------------------------------------------------------
// mHC generator (n-stream) for MI455X / gfx1250.
// One fused bandwidth-bound pass computing, per batch row b:
//     ssq[b]    = sum_d x[b,d]^2
//     dot[b,n]  = sum_d x[b,d] * w_norm[d] * Wcat[n,d]      (n = 0..23)
// via V_WMMA_F32_16X16X4_F32 (exact f32, 8 VGPRs per 16x16 accumulator tile)
// over 16-row batch tiles; tiny finalize kernel applies the RMS scale,
// affine + sigmoid / 2*sigmoid / 20-iter sinkhorn.
// Roofline: ~280 MB of mandatory traffic @ 23.3 TB/s => ~12 us floor.
// ---------------------------------------------------------------------------

#define D_TOTAL   786432      // 4*256*768
#define N_BATCH   64
#define BLOCKS_X  96
#define WAVES_PB  8
#define D_PER_WAVE (D_TOTAL / (BLOCKS_X * WAVES_PB))   // 1024

typedef __attribute__((ext_vector_type(2))) float v2f;
typedef __attribute__((ext_vector_type(4))) float v4f;
typedef __attribute__((ext_vector_type(8))) float v8f;

// ws layout (floats): [0 .. 2047] dots as [64 rows][32 n]; [2048 .. 2111] ssq
#define WS_DOTS_N 2048
#define WS_TOTAL  (WS_DOTS_N + 64)

__global__ void zero_ws_kernel(float* __restrict__ p, int n) {
    int i = blockIdx.x * blockDim.x + threadIdx.x;
    if (i < n) p[i] = 0.0f;
}

__device__ __forceinline__ v8f wmma_f32_4(v2f a, v2f b, v8f c) {
    return __builtin_amdgcn_wmma_f32_16x16x4_f32(false, a, false, b,
                                                 (short)0, c, false, false);
}

__global__ __launch_bounds__(256) void mhc_main_kernel(
    const float* __restrict__ x,      // [64, D]
    const float* __restrict__ wn,     // [D]
    const float* __restrict__ Wpre,   // [4, D]
    const float* __restrict__ Wpost,  // [4, D]
    const float* __restrict__ Wres,   // [16, D]
    float* __restrict__ ws_dots,      // [64, 32]
    float* __restrict__ ws_ss)        // [64]
{
    const int tid  = threadIdx.x;
    const int lane = tid & 31;
    const int wave = tid >> 5;
    const int tile = blockIdx.y;        // 0..3 -> batch rows tile*16 .. +15
    const int m    = lane & 15;         // A: M row in tile; B: N column
    const int h    = lane >> 4;         // K half

    const int gwave = blockIdx.x * WAVES_PB + wave;
    const int d0    = gwave * D_PER_WAVE;

    // A operand stream: row of x for this lane's M
    const float* xrow = x + (size_t)(tile * 16 + m) * D_TOTAL;

    // B operand streams: concatenated weight rows. n: 0-3 pre, 4-7 post, 8-23 res.
    const float* w0row;                 // N-tile 0: n = m (0..15)
    {
        int n = m;
        w0row = (n < 4) ? (Wpre  + (size_t)n * D_TOTAL)
              : (n < 8) ? (Wpost + (size_t)(n - 4) * D_TOTAL)
                        : (Wres  + (size_t)(n - 8) * D_TOTAL);
    }
    // N-tile 1: n = 16..31; n in [24,32) are never read downstream, so those
    // lanes just re-stream Wres row 0 and produce ignored garbage columns.
    const int n1 = 16 + m;
    const float* w1row = Wres + (size_t)((n1 < 24) ? (n1 - 8) : 0) * D_TOTAL;

    // Per-lane streaming pointers (lane half h owns a contiguous float4 per
    // 8-column logical chunk; identical K<->column remap on A and B sides).
    const float* px = xrow  + d0 + 4 * h;
    const float* pw = wn    + d0 + 4 * h;
    const float* p0 = w0row + d0 + 4 * h;
    const float* p1 = w1row + d0 + 4 * h;

    v8f c0 = {};   // dot[M][N] for N = 0..15
    v8f c1 = {};   // dot[M][N] for N = 16..31 (24..31 ignored)
    float ssq0 = 0.0f, ssq1 = 0.0f;

    for (int it = 0; it < D_PER_WAVE / 16; ++it) {
#pragma unroll
        for (int u = 0; u < 2; ++u) {
            const int o = u * 8;
            v4f ax = __builtin_nontemporal_load((const v4f*)(px + o));  // x: streamed once
            v4f wv = *(const v4f*)(pw + o);
            v4f b0 = *(const v4f*)(p0 + o);
            v4f b1 = *(const v4f*)(p1 + o);

            ssq0 = fmaf(ax.x, ax.x, ssq0);
            ssq1 = fmaf(ax.y, ax.y, ssq1);
            ssq0 = fmaf(ax.z, ax.z, ssq0);
            ssq1 = fmaf(ax.w, ax.w, ssq1);

            v4f a = ax * wv;
            v2f alo;  alo.x  = a.x;  alo.y  = a.y;
            v2f ahi;  ahi.x  = a.z;  ahi.y  = a.w;
            v2f b0lo; b0lo.x = b0.x; b0lo.y = b0.y;
            v2f b0hi; b0hi.x = b0.z; b0hi.y = b0.w;
            v2f b1lo; b1lo.x = b1.x; b1lo.y = b1.y;
            v2f b1hi; b1hi.x = b1.z; b1hi.y = b1.w;

            c0 = wmma_f32_4(alo, b0lo, c0);
            c0 = wmma_f32_4(ahi, b0hi, c0);
            c1 = wmma_f32_4(alo, b1lo, c1);
            c1 = wmma_f32_4(ahi, b1hi, c1);
        }
        px += 16; pw += 16; p0 += 16; p1 += 16;
    }
    const float ssq = ssq0 + ssq1;

    // ---- block-level reduction: C layout is (lane,v) -> M = v + 8*h, N = m
    __shared__ float lacc[16 * 32];   // [M][N]
    __shared__ float lss[16];
    for (int i = tid; i < 16 * 32; i += 256) lacc[i] = 0.0f;
    if (tid < 16) lss[tid] = 0.0f;
    __syncthreads();

#pragma unroll
    for (int v = 0; v < 8; ++v) {
        const int M = v + 8 * h;
        atomicAdd(&lacc[M * 32 + m],      c0[v]);
        atomicAdd(&lacc[M * 32 + 16 + m], c1[v]);
    }
    atomicAdd(&lss[m], ssq);
    __syncthreads();

    for (int i = tid; i < 16 * 32; i += 256)
        atomicAdd(&ws_dots[tile * 512 + i], lacc[i]);
    if (tid < 16)
        atomicAdd(&ws_ss[tile * 16 + tid], lss[tid]);
}

__global__ void mhc_finalize_kernel(
    const float* __restrict__ ws_dots,  // [64, 32]
    const float* __restrict__ ws_ss,    // [64]
    const float* __restrict__ b_pre,    // [4]
    const float* __restrict__ b_post,   // [4]
    const float* __restrict__ b_res,    // [16]
    const float* __restrict__ a_pre,    // [1]
    const float* __restrict__ a_post,   // [1]
    const float* __restrict__ a_res,    // [1]
    float* __restrict__ out)            // [1536] = pre(256) | post(256) | res(1024)
{
    const int b = threadIdx.x;
    if (b >= N_BATCH) return;

    const float ms   = ws_ss[b] / (float)D_TOTAL;
    const float invr = rsqrtf(ms + 1.1920928955078125e-7f);  // f32 eps
    const float* dp  = ws_dots + b * 32;

    const float ap  = a_pre[0];
    const float apo = a_post[0];
    const float ar  = a_res[0];

#pragma unroll
    for (int j = 0; j < 4; ++j) {
        float v = ap * invr * dp[j] + b_pre[j];
        out[b * 4 + j] = 1.0f / (1.0f + expf(-v));
    }
#pragma unroll
    for (int j = 0; j < 4; ++j) {
        float v = apo * invr * dp[4 + j] + b_post[j];
        out[256 + b * 4 + j] = 2.0f / (1.0f + expf(-v));
    }
    float R[16];
#pragma unroll
    for (int k = 0; k < 16; ++k)
        R[k] = ar * invr * dp[8 + k] + b_res[k];

#pragma unroll
    for (int i = 0; i < 4; ++i) {
        float mx = R[i * 4];
        for (int j = 1; j < 4; ++j) mx = fmaxf(mx, R[i * 4 + j]);
        for (int j = 0; j < 4; ++j) R[i * 4 + j] = expf(R[i * 4 + j] - mx);
    }
    for (int it = 0; it < 20; ++it) {
#pragma unroll
        for (int i = 0; i < 4; ++i) {
            float s = R[i*4] + R[i*4+1] + R[i*4+2] + R[i*4+3] + 1e-6f;
            float inv = 1.0f / s;
            for (int j = 0; j < 4; ++j) R[i * 4 + j] *= inv;
        }
#pragma unroll
        for (int j = 0; j < 4; ++j) {
            float s = R[j] + R[4+j] + R[8+j] + R[12+j] + 1e-6f;
            float inv = 1.0f / s;
            for (int i = 0; i < 4; ++i) R[i * 4 + j] *= inv;
        }
    }
#pragma unroll
    for (int k = 0; k < 16; ++k)
        out[512 + b * 16 + k] = R[k];
}

extern "C" void kernel_launch(void* const* d_in, const int* in_sizes, int n_in,
                              void* d_out, int out_size, void* d_ws, size_t ws_size,
                              hipStream_t stream) {
    const float* x      = (const float*)d_in[0];
    const float* wn     = (const float*)d_in[1];
    const float* Wpre   = (const float*)d_in[2];
    const float* Wpost  = (const float*)d_in[3];
    const float* Wres   = (const float*)d_in[4];
    const float* b_pre  = (const float*)d_in[5];
    const float* b_post = (const float*)d_in[6];
    const float* b_res  = (const float*)d_in[7];
    const float* a_pre  = (const float*)d_in[8];
    const float* a_post = (const float*)d_in[9];
    const float* a_res  = (const float*)d_in[10];

    float* ws      = (float*)d_ws;
    float* ws_dots = ws;
    float* ws_ss   = ws + WS_DOTS_N;

    zero_ws_kernel<<<(WS_TOTAL + 255) / 256, 256, 0, stream>>>(ws, WS_TOTAL);

    dim3 grid(BLOCKS_X, 4);
    mhc_main_kernel<<<grid, 256, 0, stream>>>(x, wn, Wpre, Wpost, Wres,
                                              ws_dots, ws_ss);

    mhc_finalize_kernel<<<1, 64, 0, stream>>>(ws_dots, ws_ss,
                                              b_pre, b_post, b_res,
                                              a_pre, a_post, a_res,
                                              (float*)d_out);
}